// FeedForwardBlock_43310450213120
// MI455X (gfx1250) — compile-verified
//
#include <hip/hip_runtime.h>
#include <hip/hip_bf16.h>
#include <math.h>

typedef float v2f __attribute__((ext_vector_type(2)));
typedef float v8f __attribute__((ext_vector_type(8)));

// ---------------- complex helpers ----------------
__device__ __forceinline__ float2 cmul(float2 a, float2 b) {
    return make_float2(a.x * b.x - a.y * b.y, a.x * b.y + a.y * b.x);
}
__device__ __forceinline__ float2 cadd(float2 a, float2 b) {
    return make_float2(a.x + b.x, a.y + b.y);
}

// =====================================================================
// Kernel 1: build the 64x64 complex matrix of the two variational layers
// (sample-independent), realified+transposed into WT[128][128] in d_ws.
// State index: wire 0 is most significant -> wire w lives at bit (5-w).
// sM layout: sM[col*64 + row] (column = input basis state).
// =====================================================================

__device__ __forceinline__ void apply_1w(float2* sM, float2 g00, float2 g01,
                                         float2 g10, float2 g11, int w) {
    const int tid = threadIdx.x;
    const int b = 5 - w;
    for (int p = tid; p < 2048; p += 256) {
        const int col  = p >> 5;
        const int rest = p & 31;
        const int low  = rest & ((1 << b) - 1);
        const int high = rest >> b;
        const int r0 = (high << (b + 1)) | low;
        const int r1 = r0 | (1 << b);
        float2 a0 = sM[col * 64 + r0];
        float2 a1 = sM[col * 64 + r1];
        sM[col * 64 + r0] = cadd(cmul(g00, a0), cmul(g01, a1));
        sM[col * 64 + r1] = cadd(cmul(g10, a0), cmul(g11, a1));
    }
    __syncthreads();
}

__device__ __forceinline__ void apply_rot(float2* sM, float phi, int w) {
    float sn, cs;
    __sincosf(phi, &sn, &cs);
    apply_1w(sM, make_float2(1.f, 0.f), make_float2(0.f, 0.f),
                 make_float2(0.f, 0.f), make_float2(cs, sn), w);
}

__device__ __forceinline__ void apply_sq(float2* sM, float r, int w) {
    const float se = 1.f / coshf(r);
    const float sr = sqrtf(se);
    apply_1w(sM, make_float2(sr, 0.f), make_float2(0.f, 0.f),
                 make_float2(0.f, 0.f), make_float2(se * sr, 0.f), w);
}

__device__ __forceinline__ void apply_disp(float2* sM, float r, int w) {
    const float pref = __expf(-0.5f * r * r);
    apply_1w(sM, make_float2(pref, 0.f), make_float2(-pref * r, 0.f),
                 make_float2(pref * r, 0.f), make_float2(pref * (1.f - r * r), 0.f), w);
}

__device__ __forceinline__ void apply_bs(float2* sM, float th, float ph, int i) {
    const float c  = cosf(th);
    const float s  = sinf(th);
    const float c2 = cosf(2.f * th);
    float epi, epr;
    __sincosf(ph, &epi, &epr);
    const int b1 = 4 - i;  // wire i+1 bit; wire i bit = b1+1 (adjacent)
    const int tid = threadIdx.x;
    for (int p = tid; p < 1024; p += 256) {
        const int col  = p >> 4;
        const int rest = p & 15;
        const int low  = rest & ((1 << b1) - 1);
        const int high = rest >> b1;
        const int base = (high << (b1 + 2)) | low;
        const int i01 = base | (1 << b1);   // n_i=0, n_{i+1}=1
        const int i10 = base | (2 << b1);   // n_i=1, n_{i+1}=0
        const int i11 = base | (3 << b1);
        float2 a01 = sM[col * 64 + i01];
        float2 a10 = sM[col * 64 + i10];
        float2 a11 = sM[col * 64 + i11];
        // new01 =  c*a01 + s*e^{+i ph}*a10 ; new10 = -s*e^{-i ph}*a01 + c*a10
        float2 n01 = cadd(make_float2(c * a01.x, c * a01.y),
                          cmul(make_float2(s * epr, s * epi), a10));
        float2 n10 = cadd(cmul(make_float2(-s * epr, s * epi), a01),
                          make_float2(c * a10.x, c * a10.y));
        sM[col * 64 + i01] = n01;
        sM[col * 64 + i10] = n10;
        sM[col * 64 + i11] = make_float2(c2 * a11.x, c2 * a11.y);
    }
    __syncthreads();
}

__global__ __launch_bounds__(256) void build_varmat_kernel(
    const float* __restrict__ var, float* __restrict__ WT) {
    __shared__ float2 sM[64 * 64];
    const int tid = threadIdx.x;
    for (int e = tid; e < 4096; e += 256) {
        const int c = e >> 6, r = e & 63;
        sM[c * 64 + r] = make_float2((r == c) ? 1.f : 0.f, 0.f);
    }
    __syncthreads();
    for (int l = 0; l < 2; ++l) {
        const float* v = var + l * 50;
        for (int i = 0; i < 5; ++i) apply_bs(sM, v[2 * i], v[2 * i + 1], i);
        for (int w = 0; w < 6; ++w) apply_rot(sM, v[10 + w], w);
        for (int w = 0; w < 6; ++w) apply_sq(sM, v[16 + w], w);
        for (int i = 0; i < 5; ++i) apply_bs(sM, v[22 + 2 * i], v[23 + 2 * i], i);
        for (int w = 0; w < 6; ++w) apply_rot(sM, v[32 + w], w);
        for (int w = 0; w < 6; ++w) apply_disp(sM, v[38 + w], w);
        for (int w = 0; w < 6; ++w) apply_rot(sM, v[44 + w], w);  // Kerr == rot at cutoff 2
    }
    // Realify + transpose:  out[j]      = sum_k  Mr[j][k] ur[k] - Mi[j][k] ui[k]
    //                       out[j+64]   = sum_k  Mi[j][k] ur[k] + Mr[j][k] ui[k]
    // WT[k][j] layout so GEMM is  amp[s][j] = sum_k u[s][k] * WT[k][j]
    for (int e = tid; e < 4096; e += 256) {
        const int k = e >> 6, j = e & 63;
        const float2 m = sM[k * 64 + j];  // M[row=j][col=k]
        WT[k * 128 + j]              =  m.x;
        WT[k * 128 + j + 64]         =  m.y;
        WT[(k + 64) * 128 + j]       = -m.y;
        WT[(k + 64) * 128 + j + 64]  =  m.x;
    }
}

// =====================================================================
// Kernel 2: fused encode -> complex matvec (f32 WMMA) -> probs -> +x -> LN
// 32 samples per 256-thread block (8 waves). Each wave owns one 16-wide
// output-column tile; 2 sample tiles of 16 -> 2 f32 accumulators.
// =====================================================================

#define SPB 32      // samples per block
#define UP 129      // LDS pitch for 128-float rows (conflict padding)
#define XP 65       // LDS pitch for 64-float x rows

__global__ __launch_bounds__(256) void quantum_ffn_kernel(
    const float* __restrict__ x, const float* __restrict__ WT,
    const float* __restrict__ gamma, const float* __restrict__ beta,
    float* __restrict__ out, int nsamp) {
    __shared__ float sU[SPB * UP];  // [Re vec(64) | Im vec(64)] per sample
    __shared__ float sO[SPB * UP];  // [Re amp(64) | Im amp(64)] per sample
    __shared__ float sX[SPB * XP];  // x features per sample

    const int tid = threadIdx.x;
    const int s0g = blockIdx.x * SPB;

    // ---- phase 0: stage x tile ----
    for (int e = tid; e < SPB * 64; e += 256) {
        const int s = e >> 6, j = e & 63;
        const int gs = s0g + s;
        sX[s * XP + j] = (gs < nsamp) ? x[(size_t)gs * 64 + j] : 0.f;
    }
    __syncthreads();

    // ---- phase 1: encoding -> product state (8 threads per sample) ----
    {
        const int s = tid >> 3;       // 0..31
        const int q = tid & 7;        // octant of the 64 basis states
        const float* X = &sX[s * XP];
        float ar[6], br[6], bi[6];
#pragma unroll
        for (int w = 0; w < 6; ++w) {
            const float s1   = 1.f / coshf(X[2 * w]);          // first squeeze
            const float rd   = X[28 + 2 * w];                  // displacement r
            const float pref = __expf(-0.5f * rd * rd);
            const float th   = X[29 + 2 * w] + X[40 + w] + X[58 + w];  // disp phase + Kerr + rot
            const float s2   = 1.f / coshf(X[46 + 2 * w]);     // second squeeze
            const float com  = sqrtf(s1) * pref;
            ar[w] = com * sqrtf(s2);
            const float mag = com * rd * s2 * sqrtf(s2);
            float sn, cs;
            __sincosf(th, &sn, &cs);
            br[w] = mag * cs;
            bi[w] = mag * sn;
        }
#pragma unroll
        for (int t = 0; t < 8; ++t) {
            const int k = q * 8 + t;   // basis index; wire w occupation = bit (5-w)
            float re = 1.f, im = 0.f;
#pragma unroll
            for (int w = 0; w < 6; ++w) {
                if ((k >> (5 - w)) & 1) {
                    const float nr = re * br[w] - im * bi[w];
                    const float ni = re * bi[w] + im * br[w];
                    re = nr; im = ni;
                } else {
                    re *= ar[w]; im *= ar[w];
                }
            }
            sU[s * UP + k]      = re;
            sU[s * UP + 64 + k] = im;
        }
    }
    __syncthreads();

    // ---- phase 2: amp = u @ WT  via V_WMMA_F32_16X16X4_F32 ----
    {
        const int lane = tid & 31;
        const int wave = tid >> 5;       // 0..7 -> output column tile
        const int n = lane & 15;         // A: M row / B: N col
        const int h = lane >> 4;         // lane half
        const int col = wave * 16 + n;
        v8f acc0 = {0.f, 0.f, 0.f, 0.f, 0.f, 0.f, 0.f, 0.f};
        v8f acc1 = {0.f, 0.f, 0.f, 0.f, 0.f, 0.f, 0.f, 0.f};
        const float* u0 = &sU[(0 * 16 + n) * UP];
        const float* u1 = &sU[(1 * 16 + n) * UP];
#pragma unroll 4
        for (int kb = 0; kb < 128; kb += 4) {
            const int k0 = kb + 2 * h;   // K = vgpr + 2*half (ISA f32 A/B layout)
            v2f bf;                      // B[k][col], K striped like A
            bf.x = WT[(k0)     * 128 + col];
            bf.y = WT[(k0 + 1) * 128 + col];
            v2f a0, a1;
            a0.x = u0[k0]; a0.y = u0[k0 + 1];
            a1.x = u1[k0]; a1.y = u1[k0 + 1];
            acc0 = __builtin_amdgcn_wmma_f32_16x16x4_f32(
                false, a0, false, bf, (short)0, acc0, false, false);
            acc1 = __builtin_amdgcn_wmma_f32_16x16x4_f32(
                false, a1, false, bf, (short)0, acc1, false, false);
        }
        // C layout: VGPR g, lane l -> M = g + 8*(l/16), N = l%16
#pragma unroll
        for (int g = 0; g < 8; ++g) {
            sO[(0 * 16 + g + 8 * h) * UP + col] = acc0[g];
            sO[(1 * 16 + g + 8 * h) * UP + col] = acc1[g];
        }
    }
    __syncthreads();

    // ---- phase 3: probs + residual + LayerNorm (8 threads per sample) ----
    {
        const int s = tid >> 3;
        const int q = tid & 7;
        const int gs = s0g + s;
        float y[8];
        float sum = 0.f, sumsq = 0.f;
#pragma unroll
        for (int t = 0; t < 8; ++t) {
            const int j = q * 8 + t;
            const float re = sO[s * UP + j];
            const float im = sO[s * UP + 64 + j];
            const float p = re * re + im * im;
            const float yy = p + sX[s * XP + j];
            y[t] = yy;
            sum += yy;
            sumsq += yy * yy;
        }
        sum   += __shfl_xor(sum, 1);  sumsq += __shfl_xor(sumsq, 1);
        sum   += __shfl_xor(sum, 2);  sumsq += __shfl_xor(sumsq, 2);
        sum   += __shfl_xor(sum, 4);  sumsq += __shfl_xor(sumsq, 4);
        const float mu  = sum * (1.f / 64.f);
        const float var = sumsq * (1.f / 64.f) - mu * mu;
        const float inv = rsqrtf(var + 1e-5f);
        if (gs < nsamp) {
#pragma unroll
            for (int t = 0; t < 8; ++t) {
                const int j = q * 8 + t;
                out[(size_t)gs * 64 + j] = (y[t] - mu) * inv * gamma[j] + beta[j];
            }
        }
    }
}

extern "C" void kernel_launch(void* const* d_in, const int* in_sizes, int n_in,
                              void* d_out, int out_size, void* d_ws, size_t ws_size,
                              hipStream_t stream) {
    const float* x     = (const float*)d_in[0];   // (B,S,64) f32
    const float* var   = (const float*)d_in[1];   // (2,50)   f32
    const float* gamma = (const float*)d_in[2];   // (64,)    f32
    const float* beta  = (const float*)d_in[3];   // (64,)    f32
    float* out = (float*)d_out;
    float* WT  = (float*)d_ws;                    // 128*128 f32 = 64 KB

    const int nsamp = in_sizes[0] / 64;

    build_varmat_kernel<<<1, 256, 0, stream>>>(var, WT);

    const int blocks = (nsamp + SPB - 1) / SPB;
    quantum_ffn_kernel<<<blocks, 256, 0, stream>>>(x, WT, gamma, beta, out, nsamp);
}